// MeshPool_37641093382406
// MI455X (gfx1250) — compile-verified
//
#include <hip/hip_runtime.h>
#include <math.h>

// ---------------- constants from the reference ----------------
#define NUM_GRAPHS 8
#define TARGETK    20000

typedef __attribute__((ext_vector_type(2))) float v2f;
typedef __attribute__((ext_vector_type(8))) float v8f;

// ---------------------------------------------------------------------------
// Kernel 1: per-edge L2 norm via V_WMMA_F32_16X16X4_F32.
// Each wave computes 16 row norms: Gram block = A * A^T accumulated over K.
// The ISA f32 A(16x4) / B(4x16) register layouts make the SAME lane data
// valid for both operands (lane l<16: row l, K pair {0,1}; lane l>=16:
// row l-16, K pair {2,3}). Diagonal of D = squared norms.
// ---------------------------------------------------------------------------
__device__ __forceinline__ float pick8(v8f a, int j) {
  float v = a[0];
  v = (j == 1) ? a[1] : v;
  v = (j == 2) ? a[2] : v;
  v = (j == 3) ? a[3] : v;
  v = (j == 4) ? a[4] : v;
  v = (j == 5) ? a[5] : v;
  v = (j == 6) ? a[6] : v;
  v = (j == 7) ? a[7] : v;
  return v;
}

__global__ __launch_bounds__(256) void score_wmma_kernel(
    const float* __restrict__ x, float* __restrict__ scores, int E, int C) {
  const int lane = threadIdx.x & 31;
  const int wave = blockIdx.x * (blockDim.x >> 5) + (threadIdx.x >> 5);
  const long long rowBase = (long long)wave * 16;
  if (rowBase >= E) return;  // uniform per wave: EXEC stays all-ones for WMMA

  const int m  = lane & 15;   // row of the 16-row tile this lane loads
  const int kh = lane >> 4;   // 0 -> K{0,1}, 1 -> K{2,3}
  long long row = rowBase + m;
  if (row >= E) row = (long long)E - 1;  // clamp (keeps all lanes active)
  const float* rp = x + row * (long long)C + 2 * kh;

  v8f acc = {};
#pragma unroll 4
  for (int k0 = 0; k0 < C; k0 += 4) {
    v2f a = *(const v2f*)(rp + k0);
    // 8 args: (neg_a, A, neg_b, B, c_mod, C, reuse_a, reuse_b)
    acc = __builtin_amdgcn_wmma_f32_16x16x4_f32(
        false, a, false, a, (short)0, acc, false, false);
  }

  // Diagonal of the 16x16 f32 D block:
  //   (m,m), m<8  -> VGPR m,   lane m      (lanes 0..7)
  //   (m,m), m>=8 -> VGPR m-8, lane m+16   (lanes 24..31)
  float v = 0.0f;
  long long orow = -1;
  if (lane < 8) {
    v = pick8(acc, lane);
    orow = rowBase + lane;
  } else if (lane >= 24) {
    v = pick8(acc, lane - 24);
    orow = rowBase + (lane - 16);
  }
  if (orow >= 0 && orow < E) scores[orow] = sqrtf(v);
}

// ---------------------------------------------------------------------------
// Kernel 2: segment boundaries. batch is sorted; segStart[g] = lower_bound(g).
// ---------------------------------------------------------------------------
__global__ void seg_bounds_kernel(const int* __restrict__ batch,
                                  int* __restrict__ segStart, int E) {
  int g = threadIdx.x;
  if (g > NUM_GRAPHS) return;
  int lo = 0, hi = E;
  while (lo < hi) {
    int mid = (lo + hi) >> 1;
    if (batch[mid] < g) lo = mid + 1; else hi = mid;
  }
  segStart[g] = lo;
}

// ---------------------------------------------------------------------------
// Kernel 3: sort keys + old2new init.
// scores >= 0, so float bits are order-monotonic; ~bits makes ascending
// radix order == descending score; stability gives lower-index tie-break.
// ---------------------------------------------------------------------------
__global__ __launch_bounds__(256) void key_init_kernel(
    const float* __restrict__ scores, unsigned* __restrict__ key,
    int* __restrict__ val, int* __restrict__ old2new, int E) {
  int i = blockIdx.x * blockDim.x + threadIdx.x;
  if (i >= E) return;
  key[i] = ~__float_as_uint(scores[i]);
  val[i] = i;
  old2new[i] = -1;
}

// ---------------------------------------------------------------------------
// Kernel 4: one stable LSD radix pass (8-bit digit), one block per graph.
// cnt[d][t] (global ws, L2-resident) is an exclusive per-thread column:
// no atomics needed, and thread-order offsets preserve stability.
// ---------------------------------------------------------------------------
__global__ __launch_bounds__(256) void sort_pass_kernel(
    const unsigned* __restrict__ keyIn, const int* __restrict__ valIn,
    unsigned* __restrict__ keyOut, int* __restrict__ valOut,
    const int* __restrict__ segStart, unsigned* __restrict__ cntG, int shift) {
  __shared__ unsigned digitBase[256];
  const int g = blockIdx.x;
  const int s = segStart[g];
  const int n = segStart[g + 1] - s;
  const int t = threadIdx.x;
  unsigned* cnt = cntG + (size_t)g * 256 * 256;

  const int chunk = (n + 255) >> 8;
  const int i0 = t * chunk;
  const int i1 = (i0 + chunk < n) ? (i0 + chunk) : n;

  // zero my column
  for (int d = 0; d < 256; ++d) cnt[d * 256 + t] = 0;
  __syncthreads();

  // count digits in my chunk (sequential stream: prefetch ~1KB ahead)
  for (int i = i0; i < i1; ++i) {
    __builtin_prefetch(&keyIn[s + i + 256], 0, 1);
    unsigned d = (keyIn[s + i] >> shift) & 255u;
    cnt[d * 256 + t] += 1u;
  }
  __syncthreads();

  // digit totals (thread t sums row t)
  unsigned tot = 0;
  for (int tt = 0; tt < 256; ++tt) tot += cnt[t * 256 + tt];
  digitBase[t] = tot;
  __syncthreads();
  if (t == 0) {
    unsigned run = 0;
    for (int d = 0; d < 256; ++d) {
      unsigned c = digitBase[d];
      digitBase[d] = run;
      run += c;
    }
  }
  __syncthreads();

  // per-(digit,thread) exclusive offsets (thread t scans row t)
  {
    unsigned run = digitBase[t];
    for (int tt = 0; tt < 256; ++tt) {
      unsigned c = cnt[t * 256 + tt];
      cnt[t * 256 + tt] = run;
      run += c;
    }
  }
  __syncthreads();

  // stable scatter
  for (int i = i0; i < i1; ++i) {
    __builtin_prefetch(&keyIn[s + i + 256], 0, 1);
    unsigned k = keyIn[s + i];
    unsigned d = (k >> shift) & 255u;
    unsigned pos = cnt[d * 256 + t]++;
    keyOut[s + pos] = k;
    valOut[s + pos] = valIn[s + i];
  }
}

// ---------------------------------------------------------------------------
// Kernel 5: build kept_idx / old2new / batch_out and gather x rows.
// float4 path: 32 lanes per row, ROWS_PER_BLK rows per 256-thread block.
// ---------------------------------------------------------------------------
#define ROWS_PER_BLK 8
__global__ __launch_bounds__(256) void scatter_gather_kernel(
    const float* __restrict__ x, const int* __restrict__ valSorted,
    const int* __restrict__ segStart, int* __restrict__ keptIdx,
    int* __restrict__ old2new, float* __restrict__ out, int C, int K, int P) {
  const int rowInBlk = threadIdx.x >> 5;   // 0..7
  const int laneC    = threadIdx.x & 31;   // float4 column index
  const int p = blockIdx.x * ROWS_PER_BLK + rowInBlk;
  if (p >= P) return;
  const int g = p / K;
  const int j = p - g * K;
  const int src = valSorted[segStart[g] + j];

  if (laneC == 0) {
    keptIdx[p] = src;
    old2new[src] = p;
    out[(size_t)P * C + (size_t)P * 4 + p] = (float)g;  // batch[kept] == g
  }
  const float4* srow = (const float4*)(x + (size_t)src * C);
  float4* drow = (float4*)(out + (size_t)p * C);
  const int nv = C >> 2;  // float4s per row (C=128 -> 32)
  for (int c = laneC; c < nv; c += 32) drow[c] = srow[c];
}

// ---------------------------------------------------------------------------
// Kernel 6: remap e2e through old2new; dropped neighbor -> self loop.
// One b128 load (int4 row) + one b128 store (float4 row) per edge.
// ---------------------------------------------------------------------------
__global__ __launch_bounds__(256) void remap_e2e_kernel(
    const int* __restrict__ e2e, const int* __restrict__ keptIdx,
    const int* __restrict__ old2new, float* __restrict__ out, int P, int C) {
  int p = blockIdx.x * blockDim.x + threadIdx.x;
  if (p >= P) return;
  const int src = keptIdx[p];
  const int4 nb4 = *(const int4*)(e2e + (size_t)src * 4);
  int nb[4] = {nb4.x, nb4.y, nb4.z, nb4.w};
  float4 r;
  float* rr = &r.x;
#pragma unroll
  for (int q = 0; q < 4; ++q) {
    int b = nb[q] < 0 ? 0 : nb[q];   // jnp.clip(e2e_kept, 0, None)
    int nn = old2new[b];
    rr[q] = (float)(nn < 0 ? p : nn);
  }
  *(float4*)(out + (size_t)P * C + (size_t)p * 4) = r;
}

// ---------------------------------------------------------------------------
extern "C" void kernel_launch(void* const* d_in, const int* in_sizes, int n_in,
                              void* d_out, int out_size, void* d_ws,
                              size_t ws_size, hipStream_t stream) {
  const float* x   = (const float*)d_in[0];
  const int* e2e   = (const int*)d_in[1];
  const int* batch = (const int*)d_in[2];

  const int E = in_sizes[2];
  const int C = in_sizes[0] / E;
  const int K = TARGETK;
  const int P = NUM_GRAPHS * K;

  // workspace layout
  char* ws = (char*)d_ws;
  size_t off = 0;
  auto take = [&](size_t bytes) -> void* {
    void* p = (void*)(ws + off);
    off += (bytes + 255) & ~(size_t)255;
    return p;
  };
  float*    scores   = (float*)take((size_t)E * 4);
  unsigned* keyA     = (unsigned*)take((size_t)E * 4);
  int*      valA     = (int*)take((size_t)E * 4);
  unsigned* keyB     = (unsigned*)take((size_t)E * 4);
  int*      valB     = (int*)take((size_t)E * 4);
  int*      old2new  = (int*)take((size_t)E * 4);
  unsigned* cntG     = (unsigned*)take((size_t)NUM_GRAPHS * 256 * 256 * 4);
  int*      segStart = (int*)take(64);
  int*      keptIdx  = (int*)take((size_t)P * 4);
  (void)ws_size; (void)n_in; (void)out_size;

  // 1) scores via FP32 WMMA (16 rows per wave, 8 waves per block)
  {
    int waves  = (E + 15) / 16;
    int blocks = (waves + 7) / 8;
    score_wmma_kernel<<<blocks, 256, 0, stream>>>(x, scores, E, C);
  }
  // 2) segment bounds
  seg_bounds_kernel<<<1, 32, 0, stream>>>(batch, segStart, E);
  // 3) keys + old2new init
  key_init_kernel<<<(E + 255) / 256, 256, 0, stream>>>(scores, keyA, valA,
                                                       old2new, E);
  // 4) four stable radix passes per graph (A->B->A->B->A)
  sort_pass_kernel<<<NUM_GRAPHS, 256, 0, stream>>>(keyA, valA, keyB, valB,
                                                   segStart, cntG, 0);
  sort_pass_kernel<<<NUM_GRAPHS, 256, 0, stream>>>(keyB, valB, keyA, valA,
                                                   segStart, cntG, 8);
  sort_pass_kernel<<<NUM_GRAPHS, 256, 0, stream>>>(keyA, valA, keyB, valB,
                                                   segStart, cntG, 16);
  sort_pass_kernel<<<NUM_GRAPHS, 256, 0, stream>>>(keyB, valB, keyA, valA,
                                                   segStart, cntG, 24);
  // 5) kept_idx / old2new / batch / x gather (float4 path)
  scatter_gather_kernel<<<(P + ROWS_PER_BLK - 1) / ROWS_PER_BLK, 256, 0,
                          stream>>>(x, valA, segStart, keptIdx, old2new,
                                    (float*)d_out, C, K, P);
  // 6) e2e remap (after old2new is complete)
  remap_e2e_kernel<<<(P + 255) / 256, 256, 0, stream>>>(e2e, keptIdx, old2new,
                                                        (float*)d_out, P, C);
}